// SpikingMNISTClassifier_55671366090810
// MI455X (gfx1250) — compile-verified
//
#include <hip/hip_runtime.h>
#include <hip/hip_bf16.h>

// ---------------------------------------------------------------------------
// Spiking MLP forward (snntorch-style LIF, 25 steps) for MI455X / gfx1250.
// GEMMs run on V_WMMA_F32_16X16X4_F32 (full fp32 fidelity; spikes are 0/1 so
// A is exact, weights stay fp32). Whole working set (~150MB) is L2-resident
// (192MB global L2), so the loop is matrix-throughput bound -> WMMA path.
// v2: 32x64 per-wave tile (MT=2 x NT=4) + double-buffered fragments so loads
// overlap WMMA issue instead of s_wait_loadcnt 0x0 before every WMMA.
// ---------------------------------------------------------------------------

typedef __attribute__((ext_vector_type(2))) float v2f;
typedef __attribute__((ext_vector_type(8))) float v8f;

#define BETA 0.9f
#define TH 1.0f
#define DT_STEPS 25
#define BATCH 1024
#define IN_F 784
#define HID 4096
#define OUTC 10
#define OUTP 16  // padded output columns

// ---------------- threefry2x32 (JAX convention, 20 rounds) -----------------
__device__ __forceinline__ unsigned rotl32(unsigned x, int n) {
  return (x << n) | (x >> (32 - n));
}

__device__ __forceinline__ void threefry2x32(unsigned k0, unsigned k1,
                                             unsigned c0, unsigned c1,
                                             unsigned& o0, unsigned& o1) {
  unsigned ks[3] = {k0, k1, k0 ^ k1 ^ 0x1BD11BDAu};
  unsigned x0 = c0 + ks[0];
  unsigned x1 = c1 + ks[1];
  const int R0[4] = {13, 15, 26, 6};
  const int R1[4] = {17, 29, 16, 24};
#pragma unroll
  for (int i = 0; i < 5; ++i) {
#pragma unroll
    for (int r = 0; r < 4; ++r) {
      int rot = (i & 1) ? R1[r] : R0[r];
      x0 += x1;
      x1 = rotl32(x1, rot);
      x1 ^= x0;
    }
    x0 += ks[(i + 1) % 3];
    x1 += ks[(i + 2) % 3] + (unsigned)(i + 1);
  }
  o0 = x0;
  o1 = x1;
}

// Poisson rate encoding for timestep t: spikes = (uniform < clip(x,0,1)).
// Reproduces jax.random.uniform(key(42), [25,1024,784]) bit-exactly.
__global__ void encode_kernel(const float* __restrict__ X,
                              float* __restrict__ Sin, int t) {
  const unsigned BF = (unsigned)BATCH * IN_F;      // 802816
  const unsigned TOTAL = (unsigned)DT_STEPS * BF;  // 20070400
  const unsigned HALF = TOTAL / 2u;                // 10035200
  unsigned idx = blockIdx.x * 256u + threadIdx.x;
  if (idx >= BF) return;
  unsigned g = (unsigned)t * BF + idx;
  unsigned c0, c1;
  bool first = (g < HALF);
  if (first) { c0 = g; c1 = g + HALF; } else { c0 = g - HALF; c1 = g; }
  unsigned o0, o1;
  threefry2x32(0u, 42u, c0, c1, o0, o1);  // key(42) -> (hi=0, lo=42)
  unsigned bits = first ? o0 : o1;
  float u = __uint_as_float((bits >> 9) | 0x3f800000u) - 1.0f;
  float p = X[idx];
  p = fminf(fmaxf(p, 0.0f), 1.0f);
  Sin[idx] = (u < p) ? 1.0f : 0.0f;
}

// Pad W2 [10,4096] -> [16,4096] with zero rows.
__global__ void padw2_kernel(const float* __restrict__ W2,
                             float* __restrict__ W2p) {
  int i = blockIdx.x * 256 + threadIdx.x;  // 16*4096 = 65536
  if (i >= OUTP * HID) return;
  int row = i >> 12;  // / 4096
  W2p[i] = (row < OUTC) ? W2[i] : 0.0f;
}

// ---------------------------------------------------------------------------
// Fused GEMM + LIF layer:
//   C = S @ W^T      (S: [1024,K] f32, W: [N,K] f32 row-major)
//   reset = (m > 1) * 1 ;  m = beta*m + C - reset ;  spike = (m > 1)
// One wave -> (16*MT) x (16*NT) tile via V_WMMA_F32_16X16X4_F32.
// A fragment: lane l holds S[row16 + (l&15)][k0 + 2*(l>>4) + {0,1}]
// B fragment: lane l holds W[n0   + (l&15)][k0 + 2*(l>>4) + {0,1}] (== B[k][n])
// C/D layout: VGPR v, lane l -> row v + 8*(l>>4), col (l&15).
// k-loop is ping-pong double buffered: next fragments load while current
// WMMAs issue (partial s_wait_loadcnt instead of full drains).
// Requires K % 8 == 0 (784, 4096 both qualify). No divergence before WMMAs.
// ---------------------------------------------------------------------------
template <int MT, int NT>
__global__ __launch_bounds__(256) void snn_layer_kernel(
    const float* __restrict__ S, const float* __restrict__ W,
    float* __restrict__ Mem, float* __restrict__ Sout,
    float* __restrict__ Out, int K, int N, int colGroups, int nValid) {
  const int lane = threadIdx.x & 31;
  const int wave = threadIdx.x >> 5;
  const int wg = blockIdx.x * 8 + wave;
  const int mBase = (wg / colGroups) * (16 * MT);
  const int nBase = (wg % colGroups) * (16 * NT);

  const int lr = lane & 15;        // row-in-tile (A) / col-in-tile (B)
  const int lk = (lane >> 4) * 2;  // k sub-offset: 0 or 2

  const float* Sp[MT];
#pragma unroll
  for (int i = 0; i < MT; ++i)
    Sp[i] = S + (size_t)(mBase + i * 16 + lr) * K + lk;
  const float* Wp[NT];
#pragma unroll
  for (int j = 0; j < NT; ++j)
    Wp[j] = W + (size_t)(nBase + j * 16 + lr) * K + lk;

  v8f acc[MT][NT];
  v8f z = {0.f, 0.f, 0.f, 0.f, 0.f, 0.f, 0.f, 0.f};
#pragma unroll
  for (int i = 0; i < MT; ++i)
#pragma unroll
    for (int j = 0; j < NT; ++j) acc[i][j] = z;

  v2f a0[MT], b0[NT], a1[MT], b1[NT];

  auto loadFrags = [&](int k, v2f* a, v2f* b) {
#pragma unroll
    for (int i = 0; i < MT; ++i) a[i] = *(const v2f*)(Sp[i] + k);
#pragma unroll
    for (int j = 0; j < NT; ++j) b[j] = *(const v2f*)(Wp[j] + k);
  };
  auto mmac = [&](const v2f* a, const v2f* b) {
#pragma unroll
    for (int i = 0; i < MT; ++i)
#pragma unroll
      for (int j = 0; j < NT; ++j)
        acc[i][j] = __builtin_amdgcn_wmma_f32_16x16x4_f32(
            false, a[i], false, b[j], (short)0, acc[i][j], false, false);
  };

  loadFrags(0, a0, b0);
  int k = 0;
  for (; k < K - 8; k += 8) {
    loadFrags(k + 4, a1, b1);  // prefetch while a0/b0 WMMAs issue
    mmac(a0, b0);
    loadFrags(k + 8, a0, b0);  // prefetch while a1/b1 WMMAs issue
    mmac(a1, b1);
  }
  // k == K - 8: two tail steps, fragments for K-8 already resident in a0/b0.
  loadFrags(K - 4, a1, b1);
  mmac(a0, b0);
  mmac(a1, b1);

  // Fused LIF epilogue in C-fragment layout.
#pragma unroll
  for (int i = 0; i < MT; ++i) {
    const int rBase = mBase + i * 16 + 8 * (lane >> 4);
#pragma unroll
    for (int j = 0; j < NT; ++j) {
      const int n = nBase + j * 16 + lr;
#pragma unroll
      for (int v = 0; v < 8; ++v) {
        const int r = rBase + v;
        const size_t idx = (size_t)r * N + n;
        float c = acc[i][j][v];
        float m = Mem[idx];
        float mn = BETA * m + c - ((m > TH) ? TH : 0.0f);
        Mem[idx] = mn;
        float sp = (mn > TH) ? 1.0f : 0.0f;
        if (Sout) Sout[idx] = sp;
        if (Out && n < nValid) Out[(size_t)r * nValid + n] += sp;
      }
    }
  }
}

extern "C" void kernel_launch(void* const* d_in, const int* in_sizes, int n_in,
                              void* d_out, int out_size, void* d_ws,
                              size_t ws_size, hipStream_t stream) {
  const float* x  = (const float*)d_in[0];  // [1024,784]
  const float* W0 = (const float*)d_in[1];  // [4096,784]
  const float* W1 = (const float*)d_in[2];  // [4096,4096]
  const float* W2 = (const float*)d_in[3];  // [10,4096]
  float* out = (float*)d_out;               // [1024,10]

  // Workspace carve-up (floats), 256B-aligned sections.
  char* ws = (char*)d_ws;
  size_t off = 0;
  auto carve = [&](size_t nfloats) {
    float* p = (float*)(ws + off);
    off += ((nfloats * sizeof(float)) + 255) & ~(size_t)255;
    return p;
  };
  float* s_in = carve((size_t)BATCH * IN_F);  // encoder output
  float* s0   = carve((size_t)BATCH * HID);   // layer0 spikes
  float* s1   = carve((size_t)BATCH * HID);   // layer1 spikes
  float* m0   = carve((size_t)BATCH * HID);   // membranes
  float* m1   = carve((size_t)BATCH * HID);
  float* m2   = carve((size_t)BATCH * OUTP);
  float* w2p  = carve((size_t)OUTP * HID);    // zero-padded W2

  // Zero state + output (graph-capture-safe async memsets).
  hipMemsetAsync(m0, 0, (size_t)BATCH * HID * sizeof(float), stream);
  hipMemsetAsync(m1, 0, (size_t)BATCH * HID * sizeof(float), stream);
  hipMemsetAsync(m2, 0, (size_t)BATCH * OUTP * sizeof(float), stream);
  hipMemsetAsync(out, 0, (size_t)BATCH * OUTC * sizeof(float), stream);

  padw2_kernel<<<(OUTP * HID + 255) / 256, 256, 0, stream>>>(W2, w2p);

  const int encBlocks = (BATCH * IN_F + 255) / 256;  // 3136
  // layers 0/1: 32 M-groups (MT=2) x 64 col-groups (NT=4) = 2048 waves = 256 blocks
  // layer  2 : 32 M-groups (MT=2) x 1 col-group  (NT=1) =   32 waves =   4 blocks
  for (int t = 0; t < DT_STEPS; ++t) {
    encode_kernel<<<encBlocks, 256, 0, stream>>>(x, s_in, t);
    snn_layer_kernel<2, 4><<<256, 256, 0, stream>>>(
        s_in, W0, m0, s0, nullptr, IN_F, HID, HID / 64, 0);
    snn_layer_kernel<2, 4><<<256, 256, 0, stream>>>(
        s0, W1, m1, s1, nullptr, HID, HID, HID / 64, 0);
    snn_layer_kernel<2, 1><<<4, 256, 0, stream>>>(
        s1, w2p, m2, nullptr, out, HID, OUTP, 1, OUTC);
  }
}